// TransE_2_73065983639795
// MI455X (gfx1250) — compile-verified
//
#include <hip/hip_runtime.h>
#include <hip/hip_bf16.h>
#include <math.h>

// Problem constants (from reference)
#define KATT      8
#define TOP_N     4
#define EMB_S     64
#define RATIO     10
#define NUM_REL   500
#define LEN_POS   1024
#define BATCH     (LEN_POS * (RATIO + 1))   // 11264
#define DIM       (TOP_N * EMB_S)           // 256
#define ENT_STRIDE (KATT * EMB_S)           // 512
#define TILE_ROWS 16
#define NUM_TILES (BATCH / TILE_ROWS)       // 704
#define LDS_ROW   260                       // padded stride (floats) -> conflict-free ds_load

typedef __attribute__((ext_vector_type(2))) float v2f;
typedef __attribute__((ext_vector_type(8))) float v8f;

// ---------------------------------------------------------------------------
// Kernel 1: per-relation precompute (only 500 relations!)
//   softmax over 8 logits, stable top-4 selection (ties -> smaller index,
//   matching jnp.argsort(-att)), pack indices into one int, store top-att sum.
// ---------------------------------------------------------------------------
__global__ void rel_prep_kernel(const float* __restrict__ rel_att,
                                int* __restrict__ top_packed,
                                float* __restrict__ s_sum) {
    int r = blockIdx.x * blockDim.x + threadIdx.x;
    if (r >= NUM_REL) return;
    float a[KATT];
#pragma unroll
    for (int i = 0; i < KATT; ++i) a[i] = rel_att[r * KATT + i];
    float mx = a[0];
#pragma unroll
    for (int i = 1; i < KATT; ++i) mx = fmaxf(mx, a[i]);
    float e[KATT];
    float tot = 0.0f;
#pragma unroll
    for (int i = 0; i < KATT; ++i) { e[i] = expf(a[i] - mx); tot += e[i]; }
    bool used[KATT];
#pragma unroll
    for (int i = 0; i < KATT; ++i) used[i] = false;
    int packed = 0;
    float tsum = 0.0f;
#pragma unroll
    for (int n = 0; n < TOP_N; ++n) {
        int   best = 0;
        float bv   = -3.0e38f;
#pragma unroll
        for (int i = 0; i < KATT; ++i) {
            if (!used[i] && e[i] > bv) { bv = e[i]; best = i; }  // strict > : stable
        }
        used[best] = true;
        packed |= best << (8 * n);
        tsum += e[best];
    }
    top_packed[r] = packed;
    s_sum[r]      = tsum / tot;
}

// ---------------------------------------------------------------------------
// Kernel 2: one wave (32 lanes) per 16 rows.
// Phase A: coalesced gather of selected head/tail segments + rel, x = h+r-t,
//          x*x staged into LDS (row stride 260 floats -> no bank conflicts).
// Phase B: 64 x V_WMMA_F32_16X16X4_F32 with B = ones accumulates
//          D[m][*] = sum_k x^2[m][k]  (exact f32 row-norm^2 for 16 rows).
// ---------------------------------------------------------------------------
__global__ __launch_bounds__(32) void transe_norm_kernel(
        const int*   __restrict__ bi,         // (BATCH,3)
        const float* __restrict__ ent,        // (NUM_ENT, 512)
        const float* __restrict__ relemb,     // (NUM_REL, 256)
        const int*   __restrict__ top_packed, // (NUM_REL)
        float*       __restrict__ out) {
    __shared__ float xsq[TILE_ROWS * LDS_ROW];   // 16,640 B

    const int lane = threadIdx.x;       // 0..31
    const int row0 = blockIdx.x * TILE_ROWS;
    const int e2   = 2 * lane;          // element pair within a 64-float segment

    // ---------------- Phase A: gather + x^2 into LDS -----------------------
    for (int m = 0; m < TILE_ROWS; ++m) {
        const int row = row0 + m;
        const int h  = bi[3 * row + 0];
        const int r  = bi[3 * row + 1];
        const int t  = bi[3 * row + 2];
        const int pk = top_packed[r];
        const float* hb = ent + (size_t)h * ENT_STRIDE;
        const float* tb = ent + (size_t)t * ENT_STRIDE;
        const float* rb = relemb + (size_t)r * DIM;
#pragma unroll
        for (int j = 0; j < TOP_N; ++j) {
            const int idx = (pk >> (8 * j)) & 0xff;
            const float2 hv = *(const float2*)(hb + idx * EMB_S + e2);
            const float2 tv = *(const float2*)(tb + idx * EMB_S + e2);
            const float2 rv = *(const float2*)(rb + j   * EMB_S + e2);
            const float x0 = hv.x + rv.x - tv.x;
            const float x1 = hv.y + rv.y - tv.y;
            float2 sq; sq.x = x0 * x0; sq.y = x1 * x1;
            *(float2*)(&xsq[m * LDS_ROW + j * EMB_S + e2]) = sq;
        }
    }
    __syncthreads();

    // ---------------- Phase B: WMMA reduction ------------------------------
    // A-matrix f32 16x4 layout: lanes 0-15 hold K=0,1 ; lanes 16-31 hold K=2,3.
    const int m   = lane & 15;
    const int off = (lane >> 4) * 2;
    v2f ones; ones.x = 1.0f; ones.y = 1.0f;
    v8f acc = {};   // zero accumulator
    for (int c = 0; c < DIM / 4; ++c) {              // 64 chunks of K=4
        const float2 a2 = *(const float2*)(&xsq[m * LDS_ROW + 4 * c + off]);
        v2f a; a.x = a2.x; a.y = a2.y;
        acc = __builtin_amdgcn_wmma_f32_16x16x4_f32(
                  /*neg_a=*/false, a, /*neg_b=*/false, ones,
                  /*c_mod=*/(short)0, acc, /*reuse_a=*/false, /*reuse_b=*/false);
    }

    // D layout: VGPR j -> lanes 0-15 = row j, lanes 16-31 = row 8+j (any N col).
    if (lane == 0 || lane == 16) {
        const int base = (lane >> 4) * 8;
#pragma unroll
        for (int j = 0; j < 8; ++j) {
            const int   row = row0 + base + j;
            const float nrm = sqrtf(acc[j]);
            if (row < LEN_POS) {
#pragma unroll
                for (int p = 0; p < RATIO; ++p)
                    out[p * LEN_POS + row] = nrm;                   // pos_norm tiled x10
            } else {
                out[RATIO * LEN_POS + (row - LEN_POS)] = nrm;       // neg_norm
            }
        }
    }
}

// ---------------------------------------------------------------------------
// Kernel 3: y = -1 fill + deterministic attention-loss reduction (one block).
// ---------------------------------------------------------------------------
__global__ void aux_kernel(const int* __restrict__ bi,
                           const float* __restrict__ s_sum,
                           float* __restrict__ out) {
    __shared__ float red[256];
    const int tid = threadIdx.x;
    const int NEG = RATIO * LEN_POS;                 // 10240
    for (int i = tid; i < NEG; i += 256)
        out[2 * NEG + i] = -1.0f;                    // y
    float acc = 0.0f;
    for (int row = tid; row < BATCH; row += 256)
        acc += 1.0f - s_sum[bi[3 * row + 1]];
    red[tid] = acc;
    __syncthreads();
    for (int s = 128; s > 0; s >>= 1) {
        if (tid < s) red[tid] += red[tid + s];
        __syncthreads();
    }
    if (tid == 0) out[3 * NEG] = red[0] / (float)BATCH;   // att_loss
}

// ---------------------------------------------------------------------------
extern "C" void kernel_launch(void* const* d_in, const int* in_sizes, int n_in,
                              void* d_out, int out_size, void* d_ws, size_t ws_size,
                              hipStream_t stream) {
    const int*   bi     = (const int*)  d_in[0];   // batch_inputs (BATCH,3) int32
    const float* ent    = (const float*)d_in[1];   // entity_embeddings
    const float* relemb = (const float*)d_in[2];   // relation_embeddings
    const float* ratt   = (const float*)d_in[3];   // rel_attention
    float* out = (float*)d_out;

    int*   top_packed = (int*)d_ws;                // 500 ints
    float* s_sum      = (float*)d_ws + 512;        // 500 floats (aligned region)

    rel_prep_kernel<<<1, 512, 0, stream>>>(ratt, top_packed, s_sum);
    transe_norm_kernel<<<NUM_TILES, 32, 0, stream>>>(bi, ent, relemb, top_packed, out);
    aux_kernel<<<1, 256, 0, stream>>>(bi, s_sum, out);
}